// GCN_20426864460069
// MI455X (gfx1250) — compile-verified
//
#include <hip/hip_runtime.h>
#include <math.h>

typedef float v2f __attribute__((ext_vector_type(2)));
typedef float v8f __attribute__((ext_vector_type(8)));

#define NUM_GRAPHS 512

static __device__ __forceinline__ void atomicAddF(float* p, float v) {
    // hardware global_atomic_add_f32 (no CAS loop)
    unsafeAtomicAdd(p, v);
}

// ---------------- utility ----------------
__global__ __launch_bounds__(256) void fill_f32(float* p, float v, int n) {
    int i = blockIdx.x * blockDim.x + threadIdx.x;
    if (i < n) p[i] = v;
}

// deg[d] += 1 per edge (deg buffer pre-filled with 1.0 for self loops)
__global__ __launch_bounds__(256) void deg_kernel(const int* __restrict__ dst, float* __restrict__ deg, int nE) {
    int e = blockIdx.x * blockDim.x + threadIdx.x;
    if (e < nE) atomicAddF(&deg[dst[e]], 1.0f);
}

// deg -> deg^{-1/2} in place
__global__ __launch_bounds__(256) void dinv_kernel(float* __restrict__ deg, int n) {
    int i = blockIdx.x * blockDim.x + threadIdx.x;
    if (i < n) {
        float d = deg[i];
        deg[i] = (d > 0.0f) ? rsqrtf(d) : 0.0f;
    }
}

// ---------------- layer 1 transform: h = x[N,3] @ W1[3,32] ----------------
__global__ __launch_bounds__(256) void gemm_3_32(const float* __restrict__ x, const float* __restrict__ W,
                                                 float* __restrict__ h, int n) {
    int idx = blockIdx.x * blockDim.x + threadIdx.x;
    if (idx >= n * 32) return;
    int i = idx >> 5, f = idx & 31;
    float x0 = x[i * 3 + 0], x1 = x[i * 3 + 1], x2 = x[i * 3 + 2];
    h[idx] = fmaf(x0, W[f], fmaf(x1, W[32 + f], x2 * W[64 + f]));
}

// ---------------- WMMA fp32 GEMM: C[rows,FOUT] = A[rows,FIN] @ W[FIN,FOUT] ----
// rows multiple of 16; FIN multiple of 4; FOUT multiple of 16 (<=64).
// One wave per 16-row tile, computing all FOUT columns.
// V_WMMA_F32_16X16X4_F32 layouts (ISA 7.12.2):
//   A 16x4 : lane(0-15)=row M, half=lane>>4 selects K pair {0,1}/{2,3}; v[0]=K even, v[1]=K odd
//   B 4x16 : lane(0-15)=col N, half selects K pair; v[0]=K even, v[1]=K odd
//   C 16x16: VGPR j -> M = j + 8*half, N = lane&15
template <int FIN, int FOUT, bool BIAS, bool RELU>
__global__ __launch_bounds__(256) void wmma_gemm(const float* __restrict__ A, const float* __restrict__ W,
                                                 const float* __restrict__ bias, float* __restrict__ C,
                                                 int rows) {
    constexpr int NT = FOUT / 16;
    int wave = (blockIdx.x * blockDim.x + threadIdx.x) >> 5;
    int lane = threadIdx.x & 31;
    int row0 = wave * 16;
    if (row0 >= rows) return;
    int m = lane & 15;
    int half = lane >> 4;

    v8f acc[NT] = {};
    const float* arow = A + (size_t)(row0 + m) * FIN + half * 2;

#pragma unroll
    for (int k0 = 0; k0 < FIN; k0 += 4) {
        v2f a;
        a.x = arow[k0 + 0];
        a.y = arow[k0 + 1];
        int krow = k0 + half * 2;
#pragma unroll
        for (int t = 0; t < NT; ++t) {
            v2f b;
            b.x = W[(krow + 0) * FOUT + t * 16 + m];
            b.y = W[(krow + 1) * FOUT + t * 16 + m];
            acc[t] = __builtin_amdgcn_wmma_f32_16x16x4_f32(false, a, false, b, (short)0, acc[t], false, false);
        }
    }

#pragma unroll
    for (int t = 0; t < NT; ++t) {
        int col = t * 16 + m;
        float bv = BIAS ? bias[col] : 0.0f;
#pragma unroll
        for (int j = 0; j < 8; ++j) {
            int r = row0 + half * 8 + j;
            float v = acc[t][j] + bv;
            if (RELU) v = fmaxf(v, 0.0f);
            C[(size_t)r * FOUT + col] = v;
        }
    }
}

// ---------------- edge scatter: agg[dst] += h[src] * dinv[src]*dinv[dst] ----
// One wave32 per edge. F==32: 1 float/lane; F==64: float2/lane.
template <int F>
__global__ __launch_bounds__(256) void edge_scatter(const float* __restrict__ h, const int* __restrict__ src,
                                                    const int* __restrict__ dst, const float* __restrict__ dinv,
                                                    float* __restrict__ agg, int nE) {
    int wave = (int)((blockIdx.x * (long long)blockDim.x + threadIdx.x) >> 5);
    int lane = threadIdx.x & 31;
    if (wave >= nE) return;
    int s = src[wave];
    int d = dst[wave];
    float nrm = dinv[s] * dinv[d];
    if (F == 32) {
        float v = h[(size_t)s * 32 + lane];
        atomicAddF(&agg[(size_t)d * 32 + lane], v * nrm);
    } else {
        const float2* hp = (const float2*)(h + (size_t)s * 64) + lane;
        float2 v = *hp;
        float* op = agg + (size_t)d * 64 + lane * 2;
        atomicAddF(op + 0, v.x * nrm);
        atomicAddF(op + 1, v.y * nrm);
    }
}

// ---------------- epilogue: out = [relu](agg + h*dinv^2 + b) ----------------
template <int F, bool RELU>
__global__ __launch_bounds__(256) void node_finish(const float* __restrict__ h, const float* __restrict__ dinv,
                                                   const float* __restrict__ b, float* __restrict__ agg, int n) {
    int idx = blockIdx.x * blockDim.x + threadIdx.x;
    if (idx >= n * F) return;
    int i = idx / F, f = idx % F;
    float di = dinv[i];
    float v = agg[idx] + h[idx] * di * di + b[f];
    agg[idx] = RELU ? fmaxf(v, 0.0f) : v;
}

// ---------------- pooling: pooled[batch[i]] += x3[i] ----------------
__global__ __launch_bounds__(256) void pool_kernel(const float* __restrict__ x3, const int* __restrict__ batch,
                                                   float* __restrict__ pooled, int n) {
    int idx = blockIdx.x * blockDim.x + threadIdx.x;
    if (idx >= n * 64) return;
    int i = idx >> 6, f = idx & 63;
    atomicAddF(&pooled[(size_t)batch[i] * 64 + f], x3[idx]);
}

// ---------------- final: z2 = z1 @ lW2 + lb2; log_softmax over graph axis ---
__global__ __launch_bounds__(512) void mlp2_logsoftmax(const float* __restrict__ z1, const float* __restrict__ lW2,
                                                       const float* __restrict__ lb2, float* __restrict__ out) {
    __shared__ float z[NUM_GRAPHS * 5];
    __shared__ float mred[5], lred[5];
    int g = threadIdx.x;  // 512 threads
    float acc[5];
#pragma unroll
    for (int c = 0; c < 5; ++c) acc[c] = lb2[c];
#pragma unroll
    for (int k = 0; k < 32; ++k) {
        float a = z1[g * 32 + k];
#pragma unroll
        for (int c = 0; c < 5; ++c) acc[c] = fmaf(a, lW2[k * 5 + c], acc[c]);
    }
#pragma unroll
    for (int c = 0; c < 5; ++c) z[g * 5 + c] = acc[c];
    __syncthreads();
    if (threadIdx.x < 5) {
        int c = threadIdx.x;
        float m = -INFINITY;
        for (int r = 0; r < NUM_GRAPHS; ++r) m = fmaxf(m, z[r * 5 + c]);
        float s = 0.0f;
        for (int r = 0; r < NUM_GRAPHS; ++r) s += __expf(z[r * 5 + c] - m);
        mred[c] = m;
        lred[c] = __logf(s);
    }
    __syncthreads();
#pragma unroll
    for (int c = 0; c < 5; ++c) out[g * 5 + c] = z[g * 5 + c] - mred[c] - lred[c];
}

static inline int cdiv(long long a, long long b) { return (int)((a + b - 1) / b); }

extern "C" void kernel_launch(void* const* d_in, const int* in_sizes, int n_in,
                              void* d_out, int out_size, void* d_ws, size_t ws_size,
                              hipStream_t stream) {
    const float* x   = (const float*)d_in[0];
    const int* ei    = (const int*)d_in[1];
    const int* batch = (const int*)d_in[2];
    const float* W1 = (const float*)d_in[3];
    const float* b1 = (const float*)d_in[4];
    const float* W2 = (const float*)d_in[5];
    const float* b2 = (const float*)d_in[6];
    const float* W3 = (const float*)d_in[7];
    const float* b3 = (const float*)d_in[8];
    const float* lW1 = (const float*)d_in[9];
    const float* lb1 = (const float*)d_in[10];
    const float* lW2 = (const float*)d_in[11];
    const float* lb2 = (const float*)d_in[12];
    float* out = (float*)d_out;

    const int n  = in_sizes[0] / 3;   // 100000 (multiple of 16)
    const int nE = in_sizes[1] / 2;   // 3200000
    const int* src = ei;
    const int* dst = ei + nE;

    // workspace carve
    float* dinv   = (float*)d_ws;                 // n
    float* B1     = dinv + n;                     // n*64
    float* B2     = B1 + (size_t)n * 64;          // n*64
    float* B3     = B2 + (size_t)n * 64;          // n*64
    float* pooled = B3 + (size_t)n * 64;          // 512*64
    float* z1     = pooled + NUM_GRAPHS * 64;     // 512*32

    const int TB = 256;

    // degree -> dinv
    fill_f32<<<cdiv(n, TB), TB, 0, stream>>>(dinv, 1.0f, n);  // self loops
    deg_kernel<<<cdiv(nE, TB), TB, 0, stream>>>(dst, dinv, nE);
    dinv_kernel<<<cdiv(n, TB), TB, 0, stream>>>(dinv, n);

    // ---- layer 1 (F=32) ----
    gemm_3_32<<<cdiv((long long)n * 32, TB), TB, 0, stream>>>(x, W1, B1, n);
    fill_f32<<<cdiv((long long)n * 32, TB), TB, 0, stream>>>(B2, 0.0f, n * 32);
    edge_scatter<32><<<cdiv((long long)nE * 32, TB), TB, 0, stream>>>(B1, src, dst, dinv, B2, nE);
    node_finish<32, true><<<cdiv((long long)n * 32, TB), TB, 0, stream>>>(B1, dinv, b1, B2, n);

    // ---- layer 2 (32 -> 64) ----
    wmma_gemm<32, 64, false, false><<<cdiv((long long)(n / 16) * 32, TB), TB, 0, stream>>>(B2, W2, nullptr, B1, n);
    fill_f32<<<cdiv((long long)n * 64, TB), TB, 0, stream>>>(B3, 0.0f, n * 64);
    edge_scatter<64><<<cdiv((long long)nE * 32, TB), TB, 0, stream>>>(B1, src, dst, dinv, B3, nE);
    node_finish<64, true><<<cdiv((long long)n * 64, TB), TB, 0, stream>>>(B1, dinv, b2, B3, n);

    // ---- layer 3 (64 -> 64, no relu) ----
    wmma_gemm<64, 64, false, false><<<cdiv((long long)(n / 16) * 32, TB), TB, 0, stream>>>(B3, W3, nullptr, B1, n);
    fill_f32<<<cdiv((long long)n * 64, TB), TB, 0, stream>>>(B2, 0.0f, n * 64);
    edge_scatter<64><<<cdiv((long long)nE * 32, TB), TB, 0, stream>>>(B1, src, dst, dinv, B2, nE);
    node_finish<64, false><<<cdiv((long long)n * 64, TB), TB, 0, stream>>>(B1, dinv, b3, B2, n);

    // ---- pooling ----
    fill_f32<<<cdiv(NUM_GRAPHS * 64, TB), TB, 0, stream>>>(pooled, 0.0f, NUM_GRAPHS * 64);
    pool_kernel<<<cdiv((long long)n * 64, TB), TB, 0, stream>>>(B2, batch, pooled, n);

    // ---- MLP head ----
    wmma_gemm<64, 32, true, true><<<cdiv((NUM_GRAPHS / 16) * 32, TB), TB, 0, stream>>>(pooled, lW1, lb1, z1, NUM_GRAPHS);
    mlp2_logsoftmax<<<1, NUM_GRAPHS, 0, stream>>>(z1, lW2, lb2, out);
}